// BOW_56075093017160
// MI455X (gfx1250) — compile-verified
//
#include <hip/hip_runtime.h>
#include <hip/hip_bf16.h>
#include <math.h>

#define VOCAB 100000
#define HIDDEN 512
#define BATCH 4096
#define TOTAL_TOKENS 524288
#define BN_EPS 1e-5f

typedef __attribute__((ext_vector_type(2))) float v2f;
typedef __attribute__((ext_vector_type(8))) float v8f;

// ---------------------------------------------------------------------------
// Kernel 1: fused gather + segment mean (EmbeddingBag, segment_ids sorted).
// One block (256 threads) per batch row; binary search for segment bounds.
// Each thread owns columns {tid, tid+256}; coalesced 512-float row reads.
// ---------------------------------------------------------------------------
__device__ __forceinline__ int lower_bound_i32(const int* __restrict__ a, int n, int v) {
    int lo = 0, hi = n;
    while (lo < hi) {
        int mid = (lo + hi) >> 1;
        if (a[mid] < v) lo = mid + 1; else hi = mid;
    }
    return lo;
}

__global__ void pool_kernel(const int* __restrict__ tokens,
                            const int* __restrict__ seg,
                            const float* __restrict__ emb,
                            float* __restrict__ pooled) {
    const int b = blockIdx.x;
    const int tid = threadIdx.x;
    __shared__ int s_bounds[2];
    if (tid == 0) {
        s_bounds[0] = lower_bound_i32(seg, TOTAL_TOKENS, b);
        s_bounds[1] = lower_bound_i32(seg, TOTAL_TOKENS, b + 1);
    }
    __syncthreads();
    const int begin = s_bounds[0];
    const int end   = s_bounds[1];

    float acc0a = 0.f, acc1a = 0.f, acc0b = 0.f, acc1b = 0.f;
    int t = begin;
    // 2x unroll: two gather rows in flight per thread
    for (; t + 1 < end; t += 2) {
        const long base0 = (long)tokens[t]     * HIDDEN;
        const long base1 = (long)tokens[t + 1] * HIDDEN;
        if (t + 8 < end) { // prefetch ahead into cache hierarchy
            __builtin_prefetch(&emb[(long)tokens[t + 8] * HIDDEN + tid], 0, 3);
        }
        acc0a += emb[base0 + tid];
        acc1a += emb[base0 + tid + 256];
        acc0b += emb[base1 + tid];
        acc1b += emb[base1 + tid + 256];
    }
    if (t < end) {
        const long base = (long)tokens[t] * HIDDEN;
        acc0a += emb[base + tid];
        acc1a += emb[base + tid + 256];
    }
    const int cnt = end - begin;
    const float inv = 1.0f / (float)(cnt > 1 ? cnt : 1);
    pooled[(long)b * HIDDEN + tid]       = (acc0a + acc0b) * inv;
    pooled[(long)b * HIDDEN + tid + 256] = (acc1a + acc1b) * inv;
}

// ---------------------------------------------------------------------------
// Kernel 2: fc1 via V_WMMA_F32_16X16X4_F32 (exact f32 matrix path).
// x[M,N] = pooled[M,K] @ W1[N,K]^T + b1,  M=4096, N=K=512.
// Block = 256 threads = 8 waves; waves tiled 2 (M) x 4 (N) -> 32x64 block tile.
// A frag: lane m=l&15, k-pair at 2*(l>>4). B frag: lane n=l&15, same k-pair.
// ---------------------------------------------------------------------------
__global__ void fc1_wmma_kernel(const float* __restrict__ pooled,
                                const float* __restrict__ W1,
                                const float* __restrict__ b1,
                                float* __restrict__ x) {
    const int lane = threadIdx.x & 31;
    const int wave = threadIdx.x >> 5;
    const int m0 = blockIdx.y * 32 + (wave >> 2) * 16;
    const int n0 = blockIdx.x * 64 + (wave & 3) * 16;

    const int mr    = lane & 15;          // row (A) / col (B) within tile
    const int kpair = (lane >> 4) * 2;    // lanes 16-31 hold K+2,K+3

    const float* __restrict__ Aptr = pooled + (long)(m0 + mr) * HIDDEN + kpair;
    const float* __restrict__ Bptr = W1     + (long)(n0 + mr) * HIDDEN + kpair; // (W1^T)[k][n] = W1[n][k]

    v8f c = {};
    #pragma unroll 4
    for (int k = 0; k < HIDDEN; k += 4) {
        v2f a = *(const v2f*)(Aptr + k);
        v2f b = *(const v2f*)(Bptr + k);
        // 8 args: (neg_a, A, neg_b, B, c_mod, C, reuse_a, reuse_b)
        c = __builtin_amdgcn_wmma_f32_16x16x4_f32(false, a, false, b,
                                                  (short)0, c, false, false);
    }

    // C/D layout: VGPR r, lanes 0-15 -> M=r, lanes 16-31 -> M=r+8; N = lane&15
    const int nc  = lane & 15;
    const int mhi = (lane >> 4) * 8;
    const float bias = b1[n0 + nc];
    #pragma unroll
    for (int r = 0; r < 8; ++r) {
        x[(long)(m0 + r + mhi) * HIDDEN + n0 + nc] = c[r] + bias;
    }
}

// ---------------------------------------------------------------------------
// Kernel 3a: per-block partial column sums / sum-of-squares (deterministic,
// no float atomics). 64 blocks x 64 rows each; thread owns cols {tid, tid+256}.
// ---------------------------------------------------------------------------
__global__ void bn_partial_kernel(const float* __restrict__ x,
                                  float* __restrict__ psum,
                                  float* __restrict__ psq) {
    const int tid = threadIdx.x;
    const int rb = blockIdx.x * 64;
    float s0 = 0.f, s1 = 0.f, q0 = 0.f, q1 = 0.f;
    for (int r = 0; r < 64; ++r) {
        const float* row = x + (long)(rb + r) * HIDDEN;
        const float v0 = row[tid];
        const float v1 = row[tid + 256];
        s0 += v0; q0 += v0 * v0;
        s1 += v1; q1 += v1 * v1;
    }
    psum[blockIdx.x * HIDDEN + tid]       = s0;
    psum[blockIdx.x * HIDDEN + tid + 256] = s1;
    psq[blockIdx.x * HIDDEN + tid]        = q0;
    psq[blockIdx.x * HIDDEN + tid + 256]  = q1;
}

// Kernel 3b: reduce 64 partials per column -> mean, invstd (biased variance)
__global__ void bn_finalize_kernel(const float* __restrict__ psum,
                                   const float* __restrict__ psq,
                                   float* __restrict__ mean,
                                   float* __restrict__ invstd) {
    const int col = blockIdx.x * blockDim.x + threadIdx.x; // 512 total
    float s = 0.f, q = 0.f;
    #pragma unroll 4
    for (int i = 0; i < 64; ++i) {
        s += psum[i * HIDDEN + col];
        q += psq[i * HIDDEN + col];
    }
    const float m   = s * (1.0f / (float)BATCH);
    const float var = q * (1.0f / (float)BATCH) - m * m;
    mean[col]   = m;
    invstd[col] = rsqrtf(var + BN_EPS);
}

// ---------------------------------------------------------------------------
// Kernel 4: fused BN-apply + ReLU + fc2 (dot with W2) + sigmoid.
// One block (256 threads) per batch row; LDS tree reduction (deterministic).
// ---------------------------------------------------------------------------
__global__ void head_kernel(const float* __restrict__ x,
                            const float* __restrict__ mean,
                            const float* __restrict__ invstd,
                            const float* __restrict__ gamma,
                            const float* __restrict__ beta,
                            const float* __restrict__ W2,
                            const float* __restrict__ b2,
                            float* __restrict__ out) {
    const int row = blockIdx.x;
    const int tid = threadIdx.x;
    const float* xr = x + (long)row * HIDDEN;

    float acc = 0.f;
    #pragma unroll
    for (int c = tid; c < HIDDEN; c += 256) {
        float h = (xr[c] - mean[c]) * invstd[c] * gamma[c] + beta[c];
        h = fmaxf(h, 0.0f);
        acc += h * W2[c];
    }

    __shared__ float sd[256];
    sd[tid] = acc;
    __syncthreads();
    for (int s = 128; s > 0; s >>= 1) {
        if (tid < s) sd[tid] += sd[tid + s];
        __syncthreads();
    }
    if (tid == 0) {
        const float v = sd[0] + b2[0];
        out[row] = 1.0f / (1.0f + expf(-v));
    }
}

// ---------------------------------------------------------------------------
extern "C" void kernel_launch(void* const* d_in, const int* in_sizes, int n_in,
                              void* d_out, int out_size, void* d_ws, size_t ws_size,
                              hipStream_t stream) {
    (void)in_sizes; (void)n_in; (void)out_size; (void)ws_size;

    const int*   tokens = (const int*)  d_in[0];
    const int*   seg    = (const int*)  d_in[1];
    const float* emb    = (const float*)d_in[2];
    const float* W1     = (const float*)d_in[3];
    const float* b1     = (const float*)d_in[4];
    const float* gamma  = (const float*)d_in[5];
    const float* beta   = (const float*)d_in[6];
    const float* W2     = (const float*)d_in[7];
    const float* b2     = (const float*)d_in[8];
    float*       out    = (float*)d_out;

    // workspace layout
    char* ws = (char*)d_ws;
    float* pooled = (float*)(ws);                                   // 4096*512 f32 = 8 MB
    float* x      = (float*)(ws + (size_t)BATCH * HIDDEN * 4);      // 8 MB
    float* psum   = (float*)(ws + (size_t)2 * BATCH * HIDDEN * 4);  // 64*512 f32
    float* psq    = psum + 64 * HIDDEN;                             // 64*512 f32
    float* mean   = psq  + 64 * HIDDEN;                             // 512 f32
    float* invstd = mean + HIDDEN;                                  // 512 f32

    // 1) gather + segment mean
    pool_kernel<<<dim3(BATCH), dim3(256), 0, stream>>>(tokens, seg, emb, pooled);

    // 2) fc1 via f32 WMMA: grid (N/64, M/32)
    fc1_wmma_kernel<<<dim3(HIDDEN / 64, BATCH / 32), dim3(256), 0, stream>>>(pooled, W1, b1, x);

    // 3) batch-norm statistics (deterministic two-pass)
    bn_partial_kernel<<<dim3(64), dim3(256), 0, stream>>>(x, psum, psq);
    bn_finalize_kernel<<<dim3(2), dim3(256), 0, stream>>>(psum, psq, mean, invstd);

    // 4) BN + ReLU + fc2 + sigmoid
    head_kernel<<<dim3(BATCH), dim3(256), 0, stream>>>(x, mean, invstd, gamma, beta, W2, b2, out);
}